// SequenceMergingSeq_16544214024448
// MI455X (gfx1250) — compile-verified
//
#include <hip/hip_runtime.h>
#include <cstdint>
#include <cstddef>

// Problem constants (match reference)
#define B_   32
#define S_   4096
#define D_   64
#define T_   64              // chunk length along S
#define NC_  (S_ / T_)       // 64 chunks per batch
#define ROWP 65              // padded LDS row stride (floats) -> conflict-free
#define EPS_DIV 1e-8f
#define EPS_LN  1e-5f

// Dynamic LDS layout (floats):
//   bufW : [0            , T_*ROWP)        decay  (overwrites W tile in place)
//   bufC : [T_*ROWP      , 2*T_*ROWP)      exp(C) then y then normed y
//   bufV : [2*T_*ROWP    , 3*T_*ROWP)      exp(C)*V (overwrites V tile)
//   sumd : [3*T_*ROWP    , +T_)
//   sume : [.. +T_       , +T_)
#define LDS_FLOATS (3 * T_ * ROWP + 2 * T_)
#define LDS_BYTES  (LDS_FLOATS * 4)

// ---------------- CDNA5 feature gates ----------------
#if defined(__AMDGCN__) && __has_builtin(__builtin_amdgcn_tensor_load_to_lds) && __has_builtin(__builtin_amdgcn_s_wait_tensorcnt)
#define HAVE_TDM 1
#else
#define HAVE_TDM 0
#endif

#if defined(__AMDGCN__) && __has_builtin(__builtin_amdgcn_wmma_f32_16x16x4_f32)
#define HAVE_WMMA 1
#else
#define HAVE_WMMA 0
#endif

typedef float v8f  __attribute__((ext_vector_type(8)));
typedef float v2f  __attribute__((ext_vector_type(2)));

#if HAVE_TDM
typedef unsigned int u32x4 __attribute__((ext_vector_type(4)));
typedef int          i32x4 __attribute__((ext_vector_type(4)));
typedef int          i32x8 __attribute__((ext_vector_type(8)));

// TDM load of a 64x64 f32 tile (rows contiguous, row stride 64 elements in
// global) into LDS at lds_byte_off, with 1 DWORD of padding inserted after
// every 64 DWORDs (-> 65-float LDS row stride, bank-conflict-free columns).
__device__ __forceinline__ void tdm_load_tile64x64(const float* gsrc, unsigned lds_byte_off) {
  unsigned long long ga = (unsigned long long)(uintptr_t)gsrc;
  u32x4 g0;
  g0.x = 1u;                                        // count=1 user descriptor
  g0.y = lds_byte_off;                              // lds_addr (bytes)
  g0.z = (unsigned)(ga & 0xFFFFFFFFull);            // global_addr[31:0]
  g0.w = ((unsigned)((ga >> 32) & 0x1FFFFFFull))    // global_addr[56:32]
         | (2u << 30);                              // type=2 ("image")
  i32x8 g1;
  // data_size=2 (4B), pad_enable=1, pad_interval=5 (64 DWORDs), pad_amount=0 (1 DWORD)
  g1[0] = (int)((2u << 16) | (1u << 20) | (5u << 22));
  g1[1] = (int)(64u << 16);   // tensor_dim0[15:0]=64 (atomic_barrier_addr=0)
  g1[2] = (int)(64u << 16);   // tensor_dim1[15:0]=64 ; tensor_dim0[31:16]=0
  g1[3] = (int)(64u << 16);   // tile_dim0=64        ; tensor_dim1[31:16]=0
  g1[4] = 64;                 // tile_dim1=64 rows   ; tile_dim2=0
  g1[5] = 64;                 // tensor_dim0_stride = 64 elements
  g1[6] = 0;
  g1[7] = 0;
  i32x4 z4 = (i32x4)0;
#if __clang_major__ >= 23
  __builtin_amdgcn_tensor_load_to_lds(g0, g1, z4, z4, (i32x8)0, 0);
#else
  __builtin_amdgcn_tensor_load_to_lds(g0, g1, z4, z4, 0);
#endif
}
#endif // HAVE_TDM

// ---------------- shared phase helpers ----------------

// Phase A: from LDS tiles of W/C/V compute decay -> bufW, exp(C) -> bufC,
// exp(C)*V -> bufV (all in place). Mapping keeps d fixed per thread.
__device__ __forceinline__ void phaseA_elementwise(float* bufW, float* bufC, float* bufV,
                                                   const float* __restrict__ tmod, int tid) {
  const float tmd = tmod[tid & (D_ - 1)];
  #pragma unroll
  for (int k = 0; k < (T_ * D_) / 256; ++k) {
    int e = tid + k * 256;
    int t = e >> 6, d = e & 63;
    int p = t * ROWP + d;
    float w  = bufW[p];
    float cc = bufC[p];
    float v  = bufV[p];
    float dec = 1.0f / (1.0f + __expf(-w * tmd));
    float ec  = __expf(cc);
    bufW[p] = dec;
    bufC[p] = ec;
    bufV[p] = ec * v;
  }
}

// Phase A2: row sums over the 64 channels for all 64 rows of bufW (->sumd)
// and bufC (->sume). WMMA path: D = A x ones[4x16] + C accumulated over K;
// every column of D holds the row sum, so B=all-ones is layout-independent.
__device__ __forceinline__ void phaseA2_rowsums(const float* bufW, const float* bufC,
                                                float* sumd, float* sume, int tid) {
#if HAVE_WMMA
  const int wv = tid >> 5;          // wave id 0..7
  const int lane = tid & 31;
  const float* src = (wv & 4) ? bufC : bufW;
  float*       dst = (wv & 4) ? sume : sumd;
  const int t0 = (wv & 3) * 16;     // 16-row tile per wave
  const int m  = lane & 15;
  const int kh = (lane >> 4) * 2;   // lanes 0-15: K=0,1 ; lanes 16-31: K=2,3
  v8f acc = (v8f)0.0f;
  v2f bones; bones.x = 1.0f; bones.y = 1.0f;
  #pragma unroll
  for (int kb = 0; kb < D_; kb += 4) {
    v2f a;
    a.x = src[(t0 + m) * ROWP + kb + kh];
    a.y = src[(t0 + m) * ROWP + kb + kh + 1];
    // (neg_a, A, neg_b, B, c_mod, C, reuse_a, reuse_b)
    acc = __builtin_amdgcn_wmma_f32_16x16x4_f32(false, a, false, bones,
                                                (short)0, acc, false, false);
  }
  // D layout: VGPR r, lanes 0-15 -> M=r (N=lane), lanes 16-31 -> M=r+8.
  // Column N=0 lives in lanes 0 and 16.
  if ((lane & 15) == 0) {
    int mbase = t0 + ((lane >> 4) ? 8 : 0);
    #pragma unroll
    for (int r = 0; r < 8; ++r) dst[mbase + r] = acc[r];
  }
#else
  if (tid < T_) {
    float sd = 0.f, se = 0.f;
    for (int d = 0; d < D_; ++d) {
      sd += bufW[tid * ROWP + d];
      se += bufC[tid * ROWP + d];
    }
    sumd[tid] = sd;
    sume[tid] = se;
  }
#endif
}

// Load the three 64x64 tiles (W, C, V) for chunk (b, c) into padded LDS.
__device__ __forceinline__ void load_tiles(const float* __restrict__ C,
                                           const float* __restrict__ V,
                                           const float* __restrict__ W,
                                           size_t base, float* smem,
                                           float* bufW, float* bufC, float* bufV,
                                           int tid) {
#if HAVE_TDM
  if (tid < 32) {  // one wave issues the TDM descriptors (SGPR-uniform)
    unsigned off = (unsigned)(uintptr_t)smem;       // LDS byte offset of dyn smem
    tdm_load_tile64x64(W + base, off + 0u);
    tdm_load_tile64x64(C + base, off + (unsigned)(T_ * ROWP * 4));
    tdm_load_tile64x64(V + base, off + (unsigned)(2 * T_ * ROWP * 4));
    __builtin_amdgcn_s_wait_tensorcnt(0);
  }
#else
  for (int k = 0; k < (T_ * D_) / 256; ++k) {
    int e = tid + k * 256;
    int t = e >> 6, d = e & 63;
    bufW[t * ROWP + d] = W[base + e];
    bufC[t * ROWP + d] = C[base + e];
    bufV[t * ROWP + d] = V[base + e];
  }
#endif
}

// ---------------- Pass 1: per-chunk local scan factors ----------------
// Per chunk (b,c): per-channel Pb[d] = prod_t decay, Qb[d] = local b-scan from 0;
// scalar Pa = prod_t mean(decay), Qa = local a-scan from 0.
__global__ void __launch_bounds__(256)
k_chunk_local(const float* __restrict__ C, const float* __restrict__ V,
              const float* __restrict__ W, const float* __restrict__ tmod,
              float* __restrict__ Pb, float* __restrict__ Qb,
              float* __restrict__ Pa, float* __restrict__ Qa) {
  extern __shared__ float smem[];
  float* bufW = smem;
  float* bufC = smem + T_ * ROWP;
  float* bufV = smem + 2 * T_ * ROWP;
  float* sumd = smem + 3 * T_ * ROWP;
  float* sume = sumd + T_;

  const int c = blockIdx.x, b = blockIdx.y;
  const int tid = threadIdx.x;
  const size_t base = ((size_t)b * S_ + (size_t)c * T_) * D_;

  load_tiles(C, V, W, base, smem, bufW, bufC, bufV, tid);
  __syncthreads();
  phaseA_elementwise(bufW, bufC, bufV, tmod, tid);
  __syncthreads();
  phaseA2_rowsums(bufW, bufC, sumd, sume, tid);
  __syncthreads();

  if (tid < D_) {
    float p = 1.f, q = 0.f;
    #pragma unroll 4
    for (int t = 0; t < T_; ++t) {
      float dec = bufW[t * ROWP + tid];
      float ecv = bufV[t * ROWP + tid];
      q = dec * q + ecv;
      p *= dec;
    }
    size_t o = ((size_t)b * NC_ + c) * D_ + tid;
    Pb[o] = p;
    Qb[o] = q;
    if (tid == 0) {
      float pa = 1.f, qa = 0.f;
      for (int t = 0; t < T_; ++t) {
        float m = sumd[t] * (1.0f / D_);
        qa = m * qa + sume[t];
        pa *= m;
      }
      size_t oc = (size_t)b * NC_ + c;
      Pa[oc] = pa;
      Qa[oc] = qa;
    }
  }
}

// ---------------- Pass 2: sequential carry combine (tiny) ----------------
__global__ void __launch_bounds__(64)
k_combine(const float* __restrict__ Pb, const float* __restrict__ Qb,
          const float* __restrict__ Pa, const float* __restrict__ Qa,
          float* __restrict__ CinA, float* __restrict__ CinB) {
  const int b = blockIdx.x;
  const int d = threadIdx.x;
  float a = 0.f, bb = 0.f;
  for (int c = 0; c < NC_; ++c) {
    size_t oc = (size_t)b * NC_ + c;
    size_t o = oc * D_ + d;
    if (d == 0) CinA[oc] = a;   // carry entering chunk c
    CinB[o] = bb;
    bb = Pb[o] * bb + Qb[o];
    a  = Pa[oc] * a + Qa[oc];   // uniform scalar recurrence (redundant per lane)
  }
}

// ---------------- Pass 3: final scan + LayerNorm + store ----------------
__global__ void __launch_bounds__(256)
k_chunk_final(const float* __restrict__ C, const float* __restrict__ V,
              const float* __restrict__ W, const float* __restrict__ tmod,
              const float* __restrict__ gamma, const float* __restrict__ beta,
              const float* __restrict__ CinA, const float* __restrict__ CinB,
              float* __restrict__ out) {
  extern __shared__ float smem[];
  float* bufW = smem;
  float* bufC = smem + T_ * ROWP;
  float* bufV = smem + 2 * T_ * ROWP;
  float* sumd = smem + 3 * T_ * ROWP;
  float* sume = sumd + T_;

  const int c = blockIdx.x, b = blockIdx.y;
  const int tid = threadIdx.x;
  const size_t base = ((size_t)b * S_ + (size_t)c * T_) * D_;

  load_tiles(C, V, W, base, smem, bufW, bufC, bufV, tid);
  __syncthreads();
  phaseA_elementwise(bufW, bufC, bufV, tmod, tid);
  __syncthreads();
  phaseA2_rowsums(bufW, bufC, sumd, sume, tid);
  __syncthreads();

  // Phase B: true-carry scan; y[t][d] = b_t / (a_t + eps) -> overwrite bufC.
  if (tid < D_) {
    size_t oc = (size_t)b * NC_ + c;
    float a  = CinA[oc];                 // uniform
    float bb = CinB[oc * D_ + tid];
    #pragma unroll 4
    for (int t = 0; t < T_; ++t) {
      float m = sumd[t] * (1.0f / D_);
      a = m * a + sume[t];               // uniform scalar recurrence
      float dec = bufW[t * ROWP + tid];
      float ecv = bufV[t * ROWP + tid];
      bb = dec * bb + ecv;
      bufC[t * ROWP + tid] = bb / (a + EPS_DIV);
    }
  }
  __syncthreads();

  // Phase C: LayerNorm per timestep (thread r owns row t=r; padded -> no conflicts).
  if (tid < T_) {
    float s = 0.f, s2 = 0.f;
    for (int d = 0; d < D_; ++d) {
      float y = bufC[tid * ROWP + d];
      s += y;
      s2 += y * y;
    }
    float mu  = s * (1.0f / D_);
    float var = s2 * (1.0f / D_) - mu * mu;
    float inv = rsqrtf(var + EPS_LN);
    for (int d = 0; d < D_; ++d) {
      float y = bufC[tid * ROWP + d];
      bufC[tid * ROWP + d] = (y - mu) * inv * gamma[d] + beta[d];
    }
  }
  __syncthreads();

  // Phase D: coalesced store.
  #pragma unroll
  for (int k = 0; k < (T_ * D_) / 256; ++k) {
    int e = tid + k * 256;
    int t = e >> 6, d = e & 63;
    out[base + e] = bufC[t * ROWP + d];
  }
}

// ---------------- host launch ----------------
extern "C" void kernel_launch(void* const* d_in, const int* in_sizes, int n_in,
                              void* d_out, int out_size, void* d_ws, size_t ws_size,
                              hipStream_t stream) {
  const float* C    = (const float*)d_in[0];
  const float* V    = (const float*)d_in[1];
  const float* W    = (const float*)d_in[2];
  const float* tmod = (const float*)d_in[3];
  const float* gam  = (const float*)d_in[4];
  const float* bet  = (const float*)d_in[5];
  float* out = (float*)d_out;

  // Workspace layout (floats). Needs ~1.6 MB.
  float* ws   = (float*)d_ws;
  const size_t nPQ = (size_t)B_ * NC_ * D_;   // 131072
  const size_t nSc = (size_t)B_ * NC_;        // 2048
  float* Pb   = ws;
  float* Qb   = Pb + nPQ;
  float* Pa   = Qb + nPQ;
  float* Qa   = Pa + nSc;
  float* CinA = Qa + nSc;
  float* CinB = CinA + nSc;
  (void)in_sizes; (void)n_in; (void)out_size; (void)ws_size;

  dim3 grid(NC_, B_);
  k_chunk_local<<<grid, 256, LDS_BYTES, stream>>>(C, V, W, tmod, Pb, Qb, Pa, Qa);
  k_combine<<<B_, 64, 0, stream>>>(Pb, Qb, Pa, Qa, CinA, CinB);
  k_chunk_final<<<grid, 256, LDS_BYTES, stream>>>(C, V, W, tmod, gam, bet, CinA, CinB, out);
}